// RiemSPD_66760971649505
// MI455X (gfx1250) — compile-verified
//
#include <hip/hip_runtime.h>

typedef __attribute__((ext_vector_type(2))) float v2f;
typedef __attribute__((ext_vector_type(8))) float v8f;

#define IN_SIZE 4096
#define HID 5
#define OUT_SIZE 2048
#define NSLICE 16
#define KSLICE (IN_SIZE / NSLICE) /* 256 */
#define EPSV 1.0e-4f

// workspace layout (floats)
#define WS_TPART 0                   // NSLICE * IN_SIZE * 16 floats
#define WS_G (NSLICE * IN_SIZE * 16) // HID * OUT_SIZE floats (G = Yp @ W2)

// ---------------------------------------------------------------------------
// Kernel 1: partial T = X @ W1 over a K-slice, via V_WMMA_F32_16X16X4_F32.
// One wave per (16-row block, K-slice). W1 slice staged in LDS padded to 16 cols.
// ---------------------------------------------------------------------------
__global__ __launch_bounds__(32) void k1_xw1(const float* __restrict__ X,
                                             const float* __restrict__ W1,
                                             float* __restrict__ ws) {
  __shared__ float ldsB[KSLICE * 16];
  const int lane = threadIdx.x;  // 0..31
  const int rb = blockIdx.x;     // row block 0..255
  const int sl = blockIdx.y;     // K slice 0..15
  const int k0 = sl * KSLICE;

  // Stage W1[k0:k0+256, 0:5] into LDS, zero-padded to 16 columns.
  for (int idx = lane; idx < KSLICE * 16; idx += 32) {
    const int row = idx >> 4;
    const int c = idx & 15;
    ldsB[idx] = (c < HID) ? W1[(k0 + row) * HID + c] : 0.0f;
  }
  __syncthreads();

  // ISA 32-bit A(16x4) layout: lanes 0-15 -> K=0,1 ; lanes 16-31 -> K=2,3
  const int m = lane & 15;   // row within tile (A) == col within tile (B)
  const int kh = lane >> 4;  // K half-select
  const float* xrow = X + (size_t)(rb * 16 + m) * IN_SIZE;

  v8f acc = {};
#pragma unroll 4
  for (int kk = 0; kk < KSLICE; kk += 4) {
    // A operand: X rows, 2 consecutive K values per lane (8B aligned)
    v2f a = *reinterpret_cast<const v2f*>(xrow + k0 + kk + 2 * kh);
    // B operand: W1 (padded) rows kk+2*kh, kk+2*kh+1, column m
    v2f b;
    b.x = ldsB[(kk + 2 * kh) * 16 + m];
    b.y = ldsB[(kk + 2 * kh + 1) * 16 + m];
    acc = __builtin_amdgcn_wmma_f32_16x16x4_f32(false, a, false, b, (short)0,
                                                acc, false, false);
  }

  // Store 16x16 partial tile.  C layout: lane l -> N = l&15, VGPR r -> M = (l>>4)*8 + r
  float* dst = ws + WS_TPART + (size_t)sl * IN_SIZE * 16;
#pragma unroll
  for (int r = 0; r < 8; ++r) {
    const int M = kh * 8 + r;
    dst[(size_t)(rb * 16 + M) * 16 + m] = acc[r];
  }
}

// ---------------------------------------------------------------------------
// Kernel 2 (1 block, 256 threads):
//   A = W1^T * Tsum  (5x5)        -> parallel reduce (registers + LDS tree)
//   B = A A^T, Jacobi eig -> U, lambda; Yp = U diag(max(sqrt(l),e)-e) U^T
//     (all 5x5 working arrays live in LDS -> zero scratch, avoids the
//      gfx1250 FI-elimination backend crash)
//   G = Yp @ W2  (5 x 2048)       -> all threads
// ---------------------------------------------------------------------------
__global__ __launch_bounds__(256) void k2_small(const float* __restrict__ W1,
                                                const float* __restrict__ W2,
                                                float* __restrict__ ws) {
  __shared__ float red[256 * 25];
  __shared__ float sM[25];  // A = W1^T X W1
  __shared__ float sB[25];  // A A^T (Jacobi working matrix)
  __shared__ float sU[25];  // accumulated eigenvectors
  __shared__ float sF[5];   // clamped-shifted spectrum
  __shared__ float sYp[25]; // Y - eps*I
  const int tid = threadIdx.x;

  float acc[25];
#pragma unroll
  for (int e = 0; e < 25; ++e) acc[e] = 0.0f;

  const float* Tp = ws + WS_TPART;
  for (int r = tid; r < IN_SIZE; r += 256) {
    float ts[HID] = {0.f, 0.f, 0.f, 0.f, 0.f};
    for (int s = 0; s < NSLICE; ++s) {
      const float* p = Tp + ((size_t)s * IN_SIZE + r) * 16;
#pragma unroll
      for (int d = 0; d < HID; ++d) ts[d] += p[d];
    }
    float w[HID];
#pragma unroll
    for (int c = 0; c < HID; ++c) w[c] = W1[r * HID + c];
#pragma unroll
    for (int c = 0; c < HID; ++c)
#pragma unroll
      for (int d = 0; d < HID; ++d) acc[c * 5 + d] += w[c] * ts[d];
  }
#pragma unroll
  for (int e = 0; e < 25; ++e) red[tid * 25 + e] = acc[e];
  __syncthreads();
  for (int s = 128; s > 0; s >>= 1) {
    if (tid < s)
      for (int e = 0; e < 25; ++e) red[tid * 25 + e] += red[(tid + s) * 25 + e];
    __syncthreads();
  }

  if (tid == 0) {
    for (int e = 0; e < 25; ++e) sM[e] = red[e];
    // B = A * A^T (symmetric PSD; eigvals = s^2 of A, eigvecs = left SVs U)
    for (int i = 0; i < 5; ++i)
      for (int j = 0; j < 5; ++j) {
        float v = 0.0f;
        for (int k = 0; k < 5; ++k) v += sM[i * 5 + k] * sM[j * 5 + k];
        sB[i * 5 + j] = v;
      }
    for (int e = 0; e < 25; ++e) sU[e] = 0.0f;
    for (int i = 0; i < 5; ++i) sU[i * 5 + i] = 1.0f;
    // Cyclic Jacobi (Golub & Van Loan): B' = J^T B J, U <- U J
    for (int sweep = 0; sweep < 12; ++sweep)
      for (int p = 0; p < 4; ++p)
        for (int q = p + 1; q < 5; ++q) {
          const float apq = sB[p * 5 + q];
          if (fabsf(apq) >= 1e-12f) {
            const float app = sB[p * 5 + p], aqq = sB[q * 5 + q];
            const float tau = (aqq - app) / (2.0f * apq);
            const float t = (tau >= 0.0f ? 1.0f : -1.0f) /
                            (fabsf(tau) + sqrtf(1.0f + tau * tau));
            const float c = 1.0f / sqrtf(1.0f + t * t);
            const float s = t * c;
            for (int k = 0; k < 5; ++k) {  // B <- B J
              const float akp = sB[k * 5 + p], akq = sB[k * 5 + q];
              sB[k * 5 + p] = c * akp - s * akq;
              sB[k * 5 + q] = s * akp + c * akq;
            }
            for (int k = 0; k < 5; ++k) {  // B <- J^T B
              const float apk = sB[p * 5 + k], aqk = sB[q * 5 + k];
              sB[p * 5 + k] = c * apk - s * aqk;
              sB[q * 5 + k] = s * apk + c * aqk;
            }
            for (int k = 0; k < 5; ++k) {  // U <- U J
              const float ukp = sU[k * 5 + p], ukq = sU[k * 5 + q];
              sU[k * 5 + p] = c * ukp - s * ukq;
              sU[k * 5 + q] = s * ukp + c * ukq;
            }
          }
        }
    for (int i = 0; i < 5; ++i) {
      float lam = sB[i * 5 + i];
      lam = lam > 0.0f ? lam : 0.0f;
      float sv = sqrtf(lam);
      sv = sv > EPSV ? sv : EPSV;
      sF[i] = sv - EPSV;  // Yp = Y - eps*I in the same eigenbasis
    }
    for (int i = 0; i < 5; ++i)
      for (int j = 0; j < 5; ++j) {
        float v = 0.0f;
        for (int k = 0; k < 5; ++k) v += sU[i * 5 + k] * sF[k] * sU[j * 5 + k];
        sYp[i * 5 + j] = v;
      }
  }
  __syncthreads();

  // G = Yp @ W2  (5 x 2048)
  float* G = ws + WS_G;
  for (int idx = tid; idx < HID * OUT_SIZE; idx += 256) {
    const int a = idx / OUT_SIZE;
    const int j = idx % OUT_SIZE;
    float v = 0.0f;
#pragma unroll
    for (int b = 0; b < HID; ++b) v += sYp[a * 5 + b] * W2[b * OUT_SIZE + j];
    G[idx] = v;
  }
}

// ---------------------------------------------------------------------------
// Kernel 3: z = W2^T * Yp * W2 + eps*I = (W2^T) @ G + eps*I, one wave per
// 16x16 tile, two V_WMMA_F32_16X16X4_F32 (K=5 zero-padded to 8).
// ---------------------------------------------------------------------------
__global__ __launch_bounds__(32) void k3_out(const float* __restrict__ W2,
                                             const float* __restrict__ G,
                                             float* __restrict__ out) {
  const int lane = threadIdx.x;
  const int j0 = blockIdx.x * 16;
  const int i0 = blockIdx.y * 16;
  const int m = lane & 15;
  const int kh = lane >> 4;
  const int i = i0 + m;  // A-matrix row  -> row i of W2^T == W2[:, i]
  const int j = j0 + m;  // B-matrix col  -> column j of G

  v8f c = {};
  // WMMA #1: K = 0..3
  v2f a0, b0;
  a0.x = W2[(2 * kh + 0) * OUT_SIZE + i];
  a0.y = W2[(2 * kh + 1) * OUT_SIZE + i];
  b0.x = G[(2 * kh + 0) * OUT_SIZE + j];
  b0.y = G[(2 * kh + 1) * OUT_SIZE + j];
  c = __builtin_amdgcn_wmma_f32_16x16x4_f32(false, a0, false, b0, (short)0, c,
                                            false, false);
  // WMMA #2: K = 4..7 (rows 5..7 are zero padding)
  v2f a1, b1;
  a1.x = (kh == 0) ? W2[4 * OUT_SIZE + i] : 0.0f;
  a1.y = 0.0f;
  b1.x = (kh == 0) ? G[4 * OUT_SIZE + j] : 0.0f;
  b1.y = 0.0f;
  c = __builtin_amdgcn_wmma_f32_16x16x4_f32(false, a1, false, b1, (short)0, c,
                                            false, false);

#pragma unroll
  for (int r = 0; r < 8; ++r) {
    const int gi = i0 + kh * 8 + r;
    const int gj = j0 + m;
    const float v = c[r] + (gi == gj ? EPSV : 0.0f);
    out[(size_t)gi * OUT_SIZE + gj] = v;
  }
}

// ---------------------------------------------------------------------------
extern "C" void kernel_launch(void* const* d_in, const int* in_sizes, int n_in,
                              void* d_out, int out_size, void* d_ws,
                              size_t ws_size, hipStream_t stream) {
  (void)in_sizes; (void)n_in; (void)out_size; (void)ws_size;
  const float* X = (const float*)d_in[0];
  const float* W1 = (const float*)d_in[1];
  const float* W2 = (const float*)d_in[2];
  float* out = (float*)d_out;
  float* ws = (float*)d_ws;

  k1_xw1<<<dim3(IN_SIZE / 16, NSLICE), 32, 0, stream>>>(X, W1, ws);
  k2_small<<<1, 256, 0, stream>>>(W1, W2, ws);
  k3_out<<<dim3(OUT_SIZE / 16, OUT_SIZE / 16), 32, 0, stream>>>(W2, ws + WS_G,
                                                                out);
}